// OctreeConv_90117003804709
// MI455X (gfx1250) — compile-verified
//
#include <hip/hip_runtime.h>

typedef __attribute__((ext_vector_type(2))) float v2f;
typedef __attribute__((ext_vector_type(4))) float v4f;
typedef __attribute__((ext_vector_type(8))) float v8f;

#define KDIM 27
#define CIN  32
#define COUT 32
#define TILE_M 64            // nodes per wave (4 sub-tiles of 16)
#define WAVES_PER_BLOCK 8    // 256 threads
#define KCHUNK 14            // k-slices of weights staged per LDS phase (<= 57KB)

__global__ __launch_bounds__(256)
void octree_conv_wmma_f32(const float* __restrict__ data,
                          const float* __restrict__ weights,
                          const int*   __restrict__ neigh_lo,  // low dwords of int64 neigh
                          float* __restrict__ out,
                          int n_tiles)
{
    __shared__ float ldsW[KCHUNK * CIN * COUT];   // 14*1024 floats = 57344 B

    const int tid  = threadIdx.x;
    const int wave = tid >> 5;
    const int lane = tid & 31;
    const int row  = lane & 15;      // M row / N col within a 16-wide tile
    const int hi   = lane >> 4;      // 0: lanes 0-15, 1: lanes 16-31

    const int tile = blockIdx.x * WAVES_PER_BLOCK + wave;   // wave-uniform
    const int base = tile * TILE_M;

    // 32-bit index bases (all fit: node*27+k < 5.4M, idx*32 < 6.4M)
    int nbase[4];
    #pragma unroll
    for (int s = 0; s < 4; ++s)
        nbase[s] = (base + s * 16 + row) * KDIM;   // element index into [N,27]

    v8f acc[4][2] = {};              // 4 M-subtiles x 2 cout halves
    const v2f zero2 = {0.0f, 0.0f};

    for (int kc = 0; kc < KDIM; kc += KCHUNK) {
        const int kcnt = (KDIM - kc) < KCHUNK ? (KDIM - kc) : KCHUNK;

        __syncthreads();  // previous phase's LDS reads done
        {   // cooperative stage: weights[kc .. kc+kcnt) -> LDS, float4 strided
            const v4f* __restrict__ wsrc =
                (const v4f*)(weights + kc * CIN * COUT);
            v4f* wdst = (v4f*)ldsW;
            const int n4 = kcnt * (CIN * COUT / 4);          // kcnt * 256
            for (int i = tid; i < n4; i += 256) wdst[i] = wsrc[i];
        }
        __syncthreads();

        if (tile < n_tiles) {   // wave-uniform branch: EXEC stays all-1s inside
            for (int kk = 0; kk < kcnt; ++kk) {
                const int k = kc + kk;

                // neighbor index per sub-tile row (lanes 16-31 duplicate 0-15);
                // int64 neigh: read low dword (sign-preserving for -1).
                // Non-temporal: neigh is streamed once, keep L2 for `data`.
                int idx[4];
                int aoff[4];                       // 32-bit element offsets
                #pragma unroll
                for (int s = 0; s < 4; ++s) {
                    idx[s] = __builtin_nontemporal_load(
                                 neigh_lo + 2 * (nbase[s] + k));
                    const int safe = idx[s] < 0 ? 0 : idx[s];
                    aoff[s] = safe * CIN + 2 * hi;
                }

                const float* __restrict__ ldsWk = ldsW + kk * (CIN * COUT);

                #pragma unroll
                for (int cb = 0; cb < 8; ++cb) {             // c_in blocks of 4
                    // A fragments: one aligned 8B load per lane, masked select
                    v2f a[4];
                    #pragma unroll
                    for (int s = 0; s < 4; ++s) {
                        v2f t = *(const v2f*)(data + aoff[s] + cb * 4);
                        a[s] = (idx[s] >= 0) ? t : zero2;
                    }

                    // B fragments from LDS: rows (cb*4 + 2*hi, +1), col c*16+row
                    const int wb = (cb * 4 + 2 * hi) * COUT + row;
                    #pragma unroll
                    for (int c = 0; c < 2; ++c) {
                        v2f b;
                        b.x = ldsWk[wb + c * 16];
                        b.y = ldsWk[wb + c * 16 + COUT];
                        #pragma unroll
                        for (int s = 0; s < 4; ++s) {
                            acc[s][c] = __builtin_amdgcn_wmma_f32_16x16x4_f32(
                                false, a[s], false, b,
                                (short)0, acc[s][c], false, false);
                        }
                    }
                }
            }
        }
    }

    if (tile < n_tiles) {
        // C/D layout: VGPR r -> row (r + 8*hi), col = lane&15 within 16-wide tile
        #pragma unroll
        for (int s = 0; s < 4; ++s) {
            #pragma unroll
            for (int c = 0; c < 2; ++c) {
                #pragma unroll
                for (int r = 0; r < 8; ++r) {
                    __builtin_nontemporal_store(
                        acc[s][c][r],
                        out + (base + s * 16 + r + 8 * hi) * COUT
                            + c * 16 + row);
                }
            }
        }
    }
}

extern "C" void kernel_launch(void* const* d_in, const int* in_sizes, int n_in,
                              void* d_out, int out_size, void* d_ws, size_t ws_size,
                              hipStream_t stream) {
    const float* data    = (const float*)d_in[0];          // [N, 32] f32
    const float* weights = (const float*)d_in[1];          // [27, 32, 32] f32
    const int*   neigh   = (const int*)d_in[2];            // [N, 27] int64 (read low dwords)
    float* out = (float*)d_out;                            // [N, 32] f32

    const int N = in_sizes[0] / CIN;                       // 200000 (multiple of 64)
    const int n_tiles = (N + TILE_M - 1) / TILE_M;         // 3125
    const int blocks = (n_tiles + WAVES_PER_BLOCK - 1) / WAVES_PER_BLOCK;

    octree_conv_wmma_f32<<<dim3(blocks), dim3(256), 0, stream>>>(
        data, weights, neigh, out, n_tiles);
}